// MultiHeadAttentionLayer_90718299226625
// MI455X (gfx1250) — compile-verified
//
#include <hip/hip_runtime.h>
#include <math.h>

typedef __attribute__((ext_vector_type(2))) float v2f;
typedef __attribute__((ext_vector_type(8))) float v8f;

#define NHEADS 8
#define ODIM   16
#define IDIM   128
#define HD     (NHEADS * ODIM)   // 128

// ---------------------------------------------------------------------------
// Zero the accumulator region of the workspace (Qsum, wV, z). d_ws is poisoned
// by the harness, so this must run every launch.
// ---------------------------------------------------------------------------
__global__ void zero_kernel(float* __restrict__ p, size_t n) {
    size_t i = (size_t)blockIdx.x * blockDim.x + threadIdx.x;
    size_t stride = (size_t)gridDim.x * blockDim.x;
    for (; i < n; i += stride) p[i] = 0.0f;
}

// ---------------------------------------------------------------------------
// QKV projection via V_WMMA_F32_16X16X4_F32.
// grid = (N/16, 3), block = 256 (8 waves). Wave w computes the 16x16 output
// tile for output columns [16w, 16w+16) of projection blockIdx.y
// (0=Q, 1=K, 2=V). Q tiles are folded (with bias) directly into Qsum[N,16]
// via atomics; K/V tiles are stored (with bias) to the workspace.
//
// A-matrix (16x4 f32, per ISA layout): lanes 0-15 = rows M, K=kk+0/kk+1;
// lanes 16-31 = rows M, K=kk+2/kk+3  -> contiguous float2 per lane from h.
// B-matrix (4x16): B[k][o] = W[o][k]  -> contiguous float2 per lane from W.
// C/D: VGPR j holds row (j + 8*hi), column lane&15.
// ---------------------------------------------------------------------------
__global__ __launch_bounds__(256) void qkv_proj_kernel(
    const float* __restrict__ h,
    const float* __restrict__ Qw, const float* __restrict__ Qb,
    const float* __restrict__ Kw, const float* __restrict__ Kb,
    const float* __restrict__ Vw, const float* __restrict__ Vb,
    float* __restrict__ Kout, float* __restrict__ Vout,
    float* __restrict__ Qsum)
{
    const int lane   = threadIdx.x & 31;
    const int waveId = threadIdx.x >> 5;        // 0..7 : head / output tile
    const int hi     = (lane >= 16) ? 1 : 0;
    const int l15    = lane & 15;
    const int proj   = blockIdx.y;              // 0=Q, 1=K, 2=V

    const float* W;
    const float* B;
    if (proj == 0)      { W = Qw; B = Qb; }
    else if (proj == 1) { W = Kw; B = Kb; }
    else                { W = Vw; B = Vb; }

    const int row0 = blockIdx.x * 16;           // node-row base of this tile
    const int ocol = waveId * 16 + l15;         // this lane's output column

    const float* aRow = h + (size_t)(row0 + l15) * IDIM + 2 * hi;
    const float* bRow = W + (size_t)ocol        * IDIM + 2 * hi;

    v8f c = {};
#pragma unroll
    for (int kk = 0; kk < IDIM; kk += 4) {
        v2f a = *(const v2f*)(aRow + kk);
        v2f b = *(const v2f*)(bRow + kk);
        // D = A*B + C  (f32, 16x16x4)
        c = __builtin_amdgcn_wmma_f32_16x16x4_f32(
                /*neg_a=*/false, a, /*neg_b=*/false, b,
                /*c_mod=*/(short)0, c, /*reuse_a=*/false, /*reuse_b=*/false);
    }

    const float bias = B[waveId * 16 + l15];

    if (proj == 0) {
        // Fold Q head-tile straight into Qsum[n, d] (sum over heads).
#pragma unroll
        for (int j = 0; j < 8; ++j) {
            const int row = row0 + j + 8 * hi;
            atomicAdd(&Qsum[(size_t)row * ODIM + l15], c[j] + bias);
        }
    } else {
        float* Out = (proj == 1) ? Kout : Vout;
#pragma unroll
        for (int j = 0; j < 8; ++j) {
            const int row = row0 + j + 8 * hi;
            Out[(size_t)row * HD + waveId * 16 + l15] = c[j] + bias;
        }
    }
}

// ---------------------------------------------------------------------------
// Edge scatter: one wave32 per edge (incl. synthetic self-loop edges).
// Lane owns 4 contiguous channels idx = lane*4 .. lane*4+3 of the 128-wide
// (head, d) space; head = lane/4. float4 loads of K[src], V[src], Qsum[dst];
// 2-step shfl_xor reduces the 16-wide dot across each 4-lane head group.
// Accumulation into wV[dst]/z[dst] via f32 global atomics (L2 resident).
// ---------------------------------------------------------------------------
__global__ __launch_bounds__(256) void edge_kernel(
    const float* __restrict__ Kf, const float* __restrict__ Vf,
    const float* __restrict__ Qsum,
    const int* __restrict__ src, const int* __restrict__ dst,
    float* __restrict__ wV, float* __restrict__ z,
    int nedges, int nnodes)
{
    const int gtid = blockIdx.x * blockDim.x + threadIdx.x;
    const int e    = gtid >> 5;                 // edge index (one per wave)
    const int lane = threadIdx.x & 31;
    if (e >= nedges + nnodes) return;

    int s, d;
    if (e < nedges) { s = src[e]; d = dst[e]; }
    else            { s = e - nedges; d = s; }  // self-loop

    const int idx = lane * 4;                   // channel base, 0..124
    const float4 k4 = *(const float4*)(Kf   + (size_t)s * HD   + idx);
    const float4 v4 = *(const float4*)(Vf   + (size_t)s * HD   + idx);
    const float4 q4 = *(const float4*)(Qsum + (size_t)d * ODIM + (idx & 15));

    float p = k4.x * q4.x + k4.y * q4.y + k4.z * q4.z + k4.w * q4.w;
    p += __shfl_xor(p, 1, 32);
    p += __shfl_xor(p, 2, 32);                  // full 16-wide dot per head

    p *= 0.25f;                                 // 1/sqrt(D), D=16
    p = fminf(5.0f, fmaxf(-5.0f, p));
    const float w = __expf(p);

    float* out = wV + (size_t)d * HD + idx;
    atomicAdd(out + 0, v4.x * w);
    atomicAdd(out + 1, v4.y * w);
    atomicAdd(out + 2, v4.z * w);
    atomicAdd(out + 3, v4.w * w);
    if ((lane & 3) == 0)
        atomicAdd(&z[(size_t)d * NHEADS + (lane >> 2)], w);
}

// ---------------------------------------------------------------------------
// Finalize: out[n, d] = mean_h( wV[n,h,d] / z[n,h] )
// ---------------------------------------------------------------------------
__global__ __launch_bounds__(256) void finalize_kernel(
    const float* __restrict__ wV, const float* __restrict__ z,
    float* __restrict__ out, int nnodes)
{
    const int tid = blockIdx.x * blockDim.x + threadIdx.x;
    if (tid >= nnodes * ODIM) return;
    const int n  = tid >> 4;
    const int dd = tid & 15;
    float acc = 0.0f;
#pragma unroll
    for (int hh = 0; hh < NHEADS; ++hh)
        acc += wV[(size_t)n * HD + hh * ODIM + dd] / z[(size_t)n * NHEADS + hh];
    out[tid] = acc * (1.0f / NHEADS);
}

// ---------------------------------------------------------------------------
extern "C" void kernel_launch(void* const* d_in, const int* in_sizes, int n_in,
                              void* d_out, int out_size, void* d_ws, size_t ws_size,
                              hipStream_t stream)
{
    const float* h   = (const float*)d_in[0];
    const float* Qw  = (const float*)d_in[1];
    const float* Qb  = (const float*)d_in[2];
    const float* Kw  = (const float*)d_in[3];
    const float* Kb  = (const float*)d_in[4];
    const float* Vw  = (const float*)d_in[5];
    const float* Vb  = (const float*)d_in[6];
    const int*   src = (const int*)d_in[7];
    const int*   dst = (const int*)d_in[8];

    const int N = in_sizes[0] / IDIM;   // 50000
    const int E = in_sizes[7];          // 1600000

    // Workspace layout (floats): K | V | Qsum | wV | z
    float* ws   = (float*)d_ws;
    float* Kbuf = ws;
    float* Vbuf = Kbuf + (size_t)N * HD;
    float* Qsum = Vbuf + (size_t)N * HD;
    float* wV   = Qsum + (size_t)N * ODIM;
    float* zb   = wV   + (size_t)N * HD;

    // 1) zero accumulators (Qsum + wV + z)
    const size_t zn = (size_t)N * (ODIM + HD + NHEADS);
    int zblocks = (int)((zn + 255) / 256);
    if (zblocks > 2048) zblocks = 2048;
    zero_kernel<<<zblocks, 256, 0, stream>>>(Qsum, zn);

    // 2) QKV projection (WMMA f32 16x16x4)
    dim3 pgrid((N + 15) / 16, 3);
    qkv_proj_kernel<<<pgrid, 256, 0, stream>>>(h, Qw, Qb, Kw, Kb, Vw, Vb,
                                               Kbuf, Vbuf, Qsum);

    // 3) edge scatter: one wave per (E + N) edges
    const long long waves   = (long long)E + N;
    const long long threads = waves * 32;
    const int eblocks = (int)((threads + 255) / 256);
    edge_kernel<<<eblocks, 256, 0, stream>>>(Kbuf, Vbuf, Qsum, src, dst,
                                             wV, zb, E, N);

    // 4) finalize
    finalize_kernel<<<(N * ODIM + 255) / 256, 256, 0, stream>>>(
        wV, zb, (float*)d_out, N);
}